// InterleavedHiddenMarkovChain_47261820125822
// MI455X (gfx1250) — compile-verified
//
#include <hip/hip_runtime.h>
#include <math.h>

// InterleavedHiddenMarkovChain — K=2, S=48, A=64, T=64, scalar logsumexp output.
// Single workgroup (768 threads = 24 wave32), fully LDS-resident.
// Final 48x48 exp-grid summation done with V_WMMA_F32_16X16X4_F32 (ones x E).

typedef float v2f __attribute__((ext_vector_type(2)));
typedef float v8f __attribute__((ext_vector_type(8)));

constexpr int KK = 2;    // number of chains
constexpr int SS = 48;   // states per chain
constexpr int AA = 64;   // alphabet
constexpr int TT = 64;   // sequence length
constexpr int SP = 49;   // padded transition row stride (coprime with 64 banks)
constexpr int DP = 65;   // padded emission/d row stride
constexpr int NTHREADS = 768;
constexpr int CELLS = SS * SS;          // 2304
constexpr int CPT = CELLS / NTHREADS;   // 3 cells per thread

__device__ __forceinline__ float lse2f(float a, float b) {
    float m = fmaxf(a, b);
    return m + log1pf(expf(-fabsf(a - b)));
}

__global__ __launch_bounds__(NTHREADS)
void ihmm_kernel(const int* __restrict__ ys,
                 const float* __restrict__ transition,
                 const float* __restrict__ emission,
                 const float* __restrict__ choice,
                 const float* __restrict__ prior,
                 float* __restrict__ out)
{
    __shared__ float s_trans[KK * SS * SP];   // 4704 f
    __shared__ float s_d[KK * SS * DP];       // 6240 f : emission, then d-table in place
    __shared__ float s_c[KK * SS];            // per-row constant
    __shared__ float s_prior[KK * SS];        // prior log-softmax
    __shared__ int   s_ys[TT];
    __shared__ float s_red[NTHREADS];
    __shared__ float s_E[SS * SP];            // exp(beta - max) grid for WMMA reduction

    const int tid = threadIdx.x;

    __builtin_prefetch(transition, 0, 0);   // global_prefetch_b8
    __builtin_prefetch(emission, 0, 0);

    // ---- Phase 1: cooperative loads into LDS (padded rows) ----
    for (int idx = tid; idx < KK * SS * SS; idx += NTHREADS) {
        int row = idx / SS, col = idx % SS;
        s_trans[row * SP + col] = transition[idx];
    }
    for (int idx = tid; idx < KK * SS * AA; idx += NTHREADS) {
        int row = idx >> 6, y = idx & 63;
        s_d[row * DP + y] = emission[idx];
    }
    if (tid < TT) s_ys[tid] = ys[tid];
    __syncthreads();

    // ---- Phase 2: row log-sum-exps ----
    const float ch0 = choice[0], ch1 = choice[1];
    const float chL = lse2f(ch0, ch1);
    if (tid < KK * SS) {                       // one thread per (i,x) row
        const int i = tid / SS;
        const int x = tid % SS;
        float m = -INFINITY;
        for (int j = 0; j < SS; ++j) m = fmaxf(m, s_trans[tid * SP + j]);
        float sum = 0.f;
        for (int j = 0; j < SS; ++j) sum += expf(s_trans[tid * SP + j] - m);
        const float transLSE = m + logf(sum);
        const float diag = s_trans[tid * SP + x];

        float me = -INFINITY;
        for (int y = 0; y < AA; ++y) me = fmaxf(me, s_d[tid * DP + y]);
        float se = 0.f;
        for (int y = 0; y < AA; ++y) se += expf(s_d[tid * DP + y] - me);
        const float emisLSE = me + logf(se);

        const float choice_l = (i == 0 ? ch0 : ch1) - chL;
        // d_i(x,y) = emission[i,x,y] + s_c[row]
        s_c[tid] = choice_l + diag - transLSE - emisLSE;
    } else if (tid < KK * SS + KK) {           // two threads: prior log-softmax rows
        const int i = tid - KK * SS;
        float m = -INFINITY;
        for (int j = 0; j < SS; ++j) m = fmaxf(m, prior[i * SS + j]);
        float sum = 0.f;
        for (int j = 0; j < SS; ++j) sum += expf(prior[i * SS + j] - m);
        const float L = m + logf(sum);
        for (int j = 0; j < SS; ++j) s_prior[i * SS + j] = prior[i * SS + j] - L;
    }
    __syncthreads();

    // ---- Phase 3: build d-table in place: d[row][y] = emission + c[row] ----
    for (int idx = tid; idx < KK * SS * AA; idx += NTHREADS) {
        int row = idx >> 6, y = idx & 63;
        s_d[row * DP + y] += s_c[row];
    }
    __syncthreads();

    // ---- Phase 4: per-cell accumulation over T steps (telescoped scan) ----
    const float pconst = lse2f(ch0 - chL, ch1 - chL);  // = LSE2(choice_l) (== 0 analytically)
    float v[CPT];
    for (int j = 0; j < CPT; ++j) {
        const int cell = tid + NTHREADS * j;
        const int x1 = cell / SS, x2 = cell % SS;
        const float* d0 = &s_d[x1 * DP];          // chain-0 row (broadcast across lanes)
        const float* d1 = &s_d[(SS + x2) * DP];   // chain-1 row (stride-65 => conflict-free)
        float g = 0.f;
        for (int t = 0; t < TT; ++t) {
            const int yt = s_ys[t];
            g += lse2f(d0[yt], d1[yt]);
        }
        v[j] = pconst + s_prior[x1] + s_prior[SS + x2] + g;
    }

    // ---- Phase 5: block max over all 2304 beta values ----
    float m = v[0];
    for (int j = 1; j < CPT; ++j) m = fmaxf(m, v[j]);
    s_red[tid] = m;
    __syncthreads();
    for (int s = 512; s > 0; s >>= 1) {
        if (tid < s && tid + s < NTHREADS)
            s_red[tid] = fmaxf(s_red[tid], s_red[tid + s]);
        __syncthreads();
    }
    const float Mx = s_red[0];

    // stage exp grid for the WMMA reduction
    for (int j = 0; j < CPT; ++j) {
        const int cell = tid + NTHREADS * j;
        const int x1 = cell / SS, x2 = cell % SS;
        s_E[x1 * SP + x2] = expf(v[j] - Mx);
    }
    __syncthreads();

    // ---- Phase 6: wave 0 sums the 48x48 grid with V_WMMA_F32_16X16X4_F32 ----
    // D = ones(16x4) x B(4x16) + C  =>  every row of D = column sums of B block.
    // A all-ones and summation over K make the result independent of the exact
    // lane<->K operand layout. EXEC is all ones for this wave (uniform branch).
    if (tid < 32) {
        const int lane = tid;
        v2f ones; ones[0] = 1.0f; ones[1] = 1.0f;
        float partial = 0.f;
        const int col_in_tile = lane & 15;
        const int half = (lane >> 4) << 1;        // 0 or 2: which pair of K-rows this lane carries
        for (int cn = 0; cn < 3; ++cn) {
            v8f acc = {0.f, 0.f, 0.f, 0.f, 0.f, 0.f, 0.f, 0.f};
            const int col = 16 * cn + col_in_tile;
            for (int ck = 0; ck < 12; ++ck) {
                const int r0 = 4 * ck + half;
                v2f b;
                b[0] = s_E[r0 * SP + col];
                b[1] = s_E[(r0 + 1) * SP + col];
                acc = __builtin_amdgcn_wmma_f32_16x16x4_f32(
                    false, ones, false, b, (short)0, acc, false, false);
            }
            // acc[0]: lanes 0-15 hold D[0][n], lanes 16-31 hold D[8][n]; both = colsum[n]
            partial += acc[0];
        }
        // full-wave sum; every column sum is present twice (M=0 and M=8 copies)
        for (int off = 16; off > 0; off >>= 1)
            partial += __shfl_xor(partial, off, 32);
        if (lane == 0) out[0] = Mx + logf(partial * 0.5f);
    }
}

extern "C" void kernel_launch(void* const* d_in, const int* in_sizes, int n_in,
                              void* d_out, int out_size, void* d_ws, size_t ws_size,
                              hipStream_t stream) {
    const int*   ys         = (const int*)  d_in[0];
    const float* transition = (const float*)d_in[1];
    const float* emission   = (const float*)d_in[2];
    const float* choice     = (const float*)d_in[3];
    const float* prior      = (const float*)d_in[4];
    float* out = (float*)d_out;
    hipLaunchKernelGGL(ihmm_kernel, dim3(1), dim3(NTHREADS), 0, stream,
                       ys, transition, emission, choice, prior, out);
}